// R2D2Head_39015482916868
// MI455X (gfx1250) — compile-verified
//
#include <hip/hip_runtime.h>

// fp32 WMMA (wave32): v_wmma_f32_16x16x4_f32
//  A: 16x4 -> v2f/lane   lanes 0-15: M=lane, K={k,k+1}; lanes 16-31: K={k+2,k+3}
//  B: 4x16 -> v2f/lane   N = lane&15, same K split
//  C/D: 16x16 -> v8f     VGPR i: M = i + 8*(lane>=16), N = lane&15
typedef float v2f  __attribute__((ext_vector_type(2)));
typedef float v8f  __attribute__((ext_vector_type(8)));
typedef unsigned int u32x4 __attribute__((ext_vector_type(4)));
typedef int   i32x4 __attribute__((ext_vector_type(4)));
typedef int   i32x8 __attribute__((ext_vector_type(8)));

#define TASK_Q 75
#define TASK_S 25
#define TASK_D 640
#define TASK_W 5
#define QCH    128                  // K-chunk of Q streamed per TDM op
#define NCH    (TASK_D / QCH)       // 5 chunks
#define SROWP  645                  // 640 + 5 pad DW (TDM pads 1 DW per 128 DW) ; 645%64=5
#define QROWP  129                  // 128 + 1 pad DW ; 129%64=1 -> conflict-free
#define WPAD   648
#define RIDGE  50.0f

// One 2D tensor_load_to_lds: tile (tile1 rows x tile0 elems) of 4-byte data,
// row stride stride0, LDS padding 1 DWORD per 128 DWORDs (bank-conflict-free rows).
__device__ __forceinline__ void tdm_load_2d(unsigned lds_byte_addr,
                                            const float* gptr,
                                            unsigned tile0, unsigned tile1,
                                            unsigned tdim0, unsigned tdim1,
                                            unsigned stride0)
{
    unsigned long long ga = (unsigned long long)(uintptr_t)gptr;
    u32x4 g0;
    g0[0] = 1u;                                   // count=1 (valid user D#)
    g0[1] = lds_byte_addr;                        // D#.lds_addr
    g0[2] = (unsigned)(ga & 0xFFFFFFFFu);         // global_addr[31:0]
    g0[3] = (unsigned)((ga >> 32) & 0x01FFFFFFu)  // global_addr[56:32]
          | (2u << 30);                           // type = 2 ("image")
    i32x8 g1;
    g1[0] = (int)((2u << 16)      // data_size = 4B
                | (1u << 20)      // pad_enable
                | (6u << 22));    // pad_interval: 128 DWORDs ; pad_amount code 0 = 1 DWORD
    g1[1] = (int)((tdim0 & 0xFFFFu) << 16);                       // tensor_dim0 lo16
    g1[2] = (int)(((tdim1 & 0xFFFFu) << 16) | ((tdim0 >> 16) & 0xFFFFu));
    g1[3] = (int)(((tile0 & 0xFFFFu) << 16) | ((tdim1 >> 16) & 0xFFFFu));
    g1[4] = (int)(tile1 & 0xFFFFu);               // tile_dim1 ; tile_dim2 = 0 (2D)
    g1[5] = (int)stride0;                         // tensor_dim0_stride lo32
    g1[6] = 0;                                    // stride0 hi16 | stride1 lo16
    g1[7] = 0;                                    // stride1 hi32 (unused, 2D)
    i32x4 z4 = {};
#if defined(__clang_major__) && (__clang_major__ >= 23)
    i32x8 z8 = {};
    __builtin_amdgcn_tensor_load_to_lds(g0, g1, z4, z4, z8, 0);
#else
    __builtin_amdgcn_tensor_load_to_lds(g0, g1, z4, z4, 0);
#endif
}

__launch_bounds__(256)
__global__ void r2d2_head_kernel(const float* __restrict__ query,
                                 const float* __restrict__ support,
                                 const int*   __restrict__ labels,
                                 float*       __restrict__ out)
{
    __shared__ float Sl[TASK_S * SROWP];       // 64,500 B  (TDM-staged support)
    __shared__ float Qb[2][TASK_Q * QROWP];    // 77,400 B  (double-buffered Q chunks)
    __shared__ float Aug[TASK_S][32];          //  3,200 B  [gram | one-hot Y | pad]
    __shared__ float Wl[TASK_W * WPAD];        // 12,960 B  W, class-major

    const int task = blockIdx.x;
    const int tid  = threadIdx.x;
    const int wave = tid >> 5;
    const int lane = tid & 31;
    const int half = lane >> 4;
    const int l16  = lane & 15;
    const int koff = half * 2;

    const float* S = support + (size_t)task * TASK_S * TASK_D;
    const float* Q = query   + (size_t)task * TASK_Q * TASK_D;

    // ---- Prologue: kick off TDM for S and Q chunk 0; zero Aug meanwhile ----
    if (wave == 0) {
        tdm_load_2d((unsigned)(uintptr_t)&Sl[0],    S, TASK_D, TASK_S,
                    TASK_D, TASK_S, TASK_D);
        tdm_load_2d((unsigned)(uintptr_t)&Qb[0][0], Q, QCH, TASK_Q,
                    TASK_D, TASK_Q, TASK_D);
    }
    for (int e = tid; e < TASK_S * 32; e += 256)
        Aug[e >> 5][e & 31] = 0.0f;
    if (wave == 0)
        __builtin_amdgcn_s_wait_tensorcnt((short)1);   // S resident (TDM in-order)
    __syncthreads();

    // ---- Gram = S*S^T + 50*I via WMMA (waves 0-3); one-hot labels (wave 4) ----
    if (wave < 4) {
        const int m0 = (wave >> 1) * 16;
        const int n0 = (wave & 1)  * 16;
        const int ra = (m0 + l16 < TASK_S) ? (m0 + l16) : (TASK_S - 1);
        const int rb = (n0 + l16 < TASK_S) ? (n0 + l16) : (TASK_S - 1);
        v8f acc = {};
        for (int k = 0; k < TASK_D; k += 4) {
            const int kk = k + koff + (k >> 7);        // TDM pad: +1 DW per 128 DW
            v2f a, b;
            a.x = Sl[ra * SROWP + kk];
            a.y = Sl[ra * SROWP + kk + 1];
            b.x = Sl[rb * SROWP + kk];                 // B = S^T
            b.y = Sl[rb * SROWP + kk + 1];
            acc = __builtin_amdgcn_wmma_f32_16x16x4_f32(
                      false, a, false, b, (short)0, acc, false, false);
        }
#pragma unroll
        for (int i = 0; i < 8; ++i) {
            int r = m0 + i + half * 8;
            int c = n0 + l16;
            if (r < TASK_S && c < TASK_S)
                Aug[r][c] = acc[i] + ((r == c) ? RIDGE : 0.0f);
        }
    } else if (wave == 4 && lane < TASK_S) {
        int lab = labels[(size_t)task * TASK_S + lane];
        if (lab < 0 || lab >= TASK_W) lab = 0;
        Aug[lane][25 + lab] = 1.0f;
    }
    __syncthreads();

    // ---- Gauss-Jordan on [gram | Y], lane = column (wave 0, lockstep LDS) ----
    if (wave == 0) {
        const int j = lane;
        for (int k = 0; k < TASK_S; ++k) {
            float pivinv = 1.0f / Aug[k][k];
            float rkj = Aug[k][j] * pivinv;
            Aug[k][j] = rkj;
            for (int i = 0; i < TASK_S; ++i) {
                if (i == k) continue;
                float aik = Aug[i][k];
                Aug[i][j] -= aik * rkj;
            }
        }
    }
    __syncthreads();

    // ---- W[c][d] = sum_s S[s][d] * X[s][c] ----
    for (int idx = tid; idx < TASK_D * TASK_W; idx += 256) {
        int c  = idx / TASK_D;
        int dc = idx - c * TASK_D;
        float acc = 0.0f;
#pragma unroll
        for (int r = 0; r < TASK_S; ++r)
            acc += Sl[r * SROWP + dc + (dc >> 7)] * Aug[r][25 + c];
        Wl[c * WPAD + dc] = acc;
    }
    __syncthreads();

    // ---- Logits = Q @ W : TDM double-buffered K-chunks + WMMA (waves 0-4) ----
    int row = wave * 16 + l16;
    if (row > TASK_Q - 1) row = TASK_Q - 1;            // clamp pad rows
    const int col = l16;
    const bool cok = (col < TASK_W);
    v8f acc = {};
    for (int c = 0; c < NCH; ++c) {
        if (wave == 0)
            __builtin_amdgcn_s_wait_tensorcnt((short)0);   // chunk c complete
        __syncthreads();                                   // publish buffer to all waves
        if (wave == 0 && c + 1 < NCH)                      // overlap next DMA w/ compute
            tdm_load_2d((unsigned)(uintptr_t)&Qb[(c + 1) & 1][0],
                        Q + (size_t)(c + 1) * QCH, QCH, TASK_Q,
                        TASK_D, TASK_Q, TASK_D);
        if (wave < 5) {
            const float* qb = &Qb[c & 1][row * QROWP];
            const int kbase = c * QCH;
            for (int kc = 0; kc < QCH; kc += 4) {
                v2f a = *(const v2f*)(qb + kc + koff);      // ds_load_b64, no conflicts
                v2f b;
                b.x = cok ? Wl[col * WPAD + kbase + kc + koff]     : 0.0f;
                b.y = cok ? Wl[col * WPAD + kbase + kc + koff + 1] : 0.0f;
                acc = __builtin_amdgcn_wmma_f32_16x16x4_f32(
                          false, a, false, b, (short)0, acc, false, false);
            }
        }
    }
#pragma unroll
    for (int i = 0; i < 8; ++i) {
        int r = wave * 16 + i + half * 8;
        int cc = l16;
        if (wave < 5 && r < TASK_Q && cc < TASK_W)
            out[((size_t)task * TASK_Q + r) * TASK_W + cc] = acc[i];
    }
}

extern "C" void kernel_launch(void* const* d_in, const int* in_sizes, int n_in,
                              void* d_out, int out_size, void* d_ws, size_t ws_size,
                              hipStream_t stream) {
    const float* query   = (const float*)d_in[0];
    const float* support = (const float*)d_in[1];
    const int*   labels  = (const int*)d_in[2];
    float*       out     = (float*)d_out;

    const int B = in_sizes[2] / TASK_S;   // 2048 tasks
    r2d2_head_kernel<<<dim3(B), dim3(256), 0, stream>>>(query, support, labels, out);
}